// TsCorr_17051020165337
// MI455X (gfx1250) — compile-verified
//
#include <hip/hip_runtime.h>
#include <math.h>

typedef __attribute__((ext_vector_type(2))) float v2f;
typedef __attribute__((ext_vector_type(8))) float v8f;

#define BATCH   256
#define NF      32
#define LEN     2048
#define WSIZE   20
#define WSTRIDE 5
#define T_OUT   406                       // (2048-20)/5 + 1
#define NPAIR   496                       // 32*31/2
#define TT      16                        // windows per workgroup
#define SPAN    ((TT - 1) * WSTRIDE + WSIZE)   // 95 columns of x per tile
#define EPSC    1e-8f

__global__ __launch_bounds__(256) void tscorr_wmma_kernel(const float* __restrict__ x,
                                                          float* __restrict__ out) {
  __shared__ float xs[NF][SPAN];          // raw x slab (pitch 95: odd -> bank-conflict-free)
  __shared__ float sm[TT][NF];            // per (window, feature) mean
  __shared__ float sn[TT][NF];            // per (window, feature) norm of centered window
  __shared__ float sout[NPAIR][TT + 1];   // staged output, +1 pad (stride 17, gcd(17,64)=1)

  const int tid  = threadIdx.x;
  const int b    = blockIdx.y;
  const int t0   = blockIdx.x * TT;
  const int nv   = min(TT, T_OUT - t0);   // valid windows in this tile
  const int base = t0 * WSTRIDE;

  // ---- Phase 1: cooperative load of the shared x slab (coalesced along L) ----
  const float* xb = x + (size_t)b * NF * LEN;
  for (int idx = tid; idx < NF * SPAN; idx += 256) {
    int f   = idx / SPAN;
    int col = idx - f * SPAN;
    int g   = base + col;
    xs[f][col] = (g < LEN) ? xb[(size_t)f * LEN + g] : 0.0f;
  }
  __syncthreads();

  // ---- Phase 2: per (window, feature) mean and centered-norm ----
  for (int idx = tid; idx < TT * NF; idx += 256) {
    int tt = idx >> 5;
    int f  = idx & 31;
    if (tt < nv) {
      int toff = tt * WSTRIDE;
      float s1 = 0.0f, s2 = 0.0f;
#pragma unroll
      for (int s = 0; s < WSIZE; ++s) {
        float v = xs[f][toff + s];
        s1 += v;
        s2 += v * v;
      }
      float m  = s1 * (1.0f / (float)WSIZE);
      float n2 = s2 - (float)WSIZE * m * m;   // sum of squares of centered window
      sm[tt][f] = m;
      sn[tt][f] = sqrtf(fmaxf(n2, 0.0f));
    }
  }
  __syncthreads();

  // ---- Phase 3: one wave per window computes the 32x32 Gram via fp32 WMMA ----
  const int lane = tid & 31;
  const int wid  = tid >> 5;
  const int r    = lane & 15;   // row (A frag) / col (B frag)
  const int h    = lane >> 4;   // lane half -> K pair select

  for (int tt = wid; tt < nv; tt += 8) {          // wave-uniform: EXEC all-1s at WMMA
    const int toff = tt * WSTRIDE;
    v8f d00 = {}; v8f d01 = {}; v8f d11 = {};
#pragma unroll
    for (int ks = 0; ks < 5; ++ks) {              // K = 20 = 5 steps of K=4
      int k = toff + ks * 4 + 2 * h;              // VGPR0=K{0|2}, VGPR1=K{1|3} by lane half
      v2f a0, a1;
      a0.x = xs[r][k];       a0.y = xs[r][k + 1];        // features 0..15
      a1.x = xs[16 + r][k];  a1.y = xs[16 + r][k + 1];   // features 16..31
      // Gram: B = A^T has the same per-lane fragment as A for fp32 16x16x4
      d00 = __builtin_amdgcn_wmma_f32_16x16x4_f32(false, a0, false, a0, (short)0, d00, false, false);
      d01 = __builtin_amdgcn_wmma_f32_16x16x4_f32(false, a0, false, a1, (short)0, d01, false, false);
      d11 = __builtin_amdgcn_wmma_f32_16x16x4_f32(false, a1, false, a1, (short)0, d11, false, false);
    }

    // D layout: VGPR v, lane -> M = v + 8*h, N = lane&15
    const int   c   = r;
    const float mjB = sm[tt][16 + c], njB = sn[tt][16 + c];
    const float mjA = sm[tt][c],      njA = sn[tt][c];
#pragma unroll
    for (int v = 0; v < 8; ++v) {
      int   i  = v + 8 * h;
      float mi = sm[tt][i], ni = sn[tt][i];
      // T01 tile: pair (i, 16+c) -- always upper triangle
      {
        float num = d01[v] - (float)WSIZE * mi * mjB;
        int   p   = (i * (63 - i)) / 2 + (16 + c - i - 1);
        sout[p][tt] = num * __builtin_amdgcn_rcpf(ni * njB + EPSC);
      }
      if (c > i) {
        // T00 tile: pair (i, c)
        float num0 = d00[v] - (float)WSIZE * mi * mjA;
        int   p0   = (i * (63 - i)) / 2 + (c - i - 1);
        sout[p0][tt] = num0 * __builtin_amdgcn_rcpf(ni * njA + EPSC);
        // T11 tile: pair (16+i, 16+c)
        float mi2 = sm[tt][16 + i], ni2 = sn[tt][16 + i];
        float num1 = d11[v] - (float)WSIZE * mi2 * mjB;
        int   i2   = 16 + i;
        int   p1   = (i2 * (63 - i2)) / 2 + (16 + c - i2 - 1);
        sout[p1][tt] = num1 * __builtin_amdgcn_rcpf(ni2 * njB + EPSC);
      }
    }
  }
  __syncthreads();

  // ---- Phase 4: write-out along contiguous t axis, paired into b64 stores ----
  // Row base (pidx*T_OUT + t0) is always even -> 8-byte aligned float2 stores.
  float* ob = out + (size_t)b * NPAIR * T_OUT + t0;
  for (int q = tid; q < NPAIR * (TT / 2); q += 256) {
    int p   = q >> 3;            // pair row
    int tt2 = (q & 7) * 2;       // even t within tile
    if (tt2 + 1 < nv) {
      float2 val;
      val.x = sout[p][tt2];
      val.y = sout[p][tt2 + 1];
      *(float2*)(ob + (size_t)p * T_OUT + tt2) = val;
    } else if (tt2 < nv) {
      ob[(size_t)p * T_OUT + tt2] = sout[p][tt2];
    }
  }
}

extern "C" void kernel_launch(void* const* d_in, const int* in_sizes, int n_in,
                              void* d_out, int out_size, void* d_ws, size_t ws_size,
                              hipStream_t stream) {
  (void)in_sizes; (void)n_in; (void)d_ws; (void)ws_size; (void)out_size;
  const float* x  = (const float*)d_in[0];
  float*       out = (float*)d_out;
  dim3 grid((T_OUT + TT - 1) / TT, BATCH);   // 26 x 256 workgroups
  tscorr_wmma_kernel<<<grid, 256, 0, stream>>>(x, out);
}